// AdaptiveCombiner_29583734735132
// MI455X (gfx1250) — compile-verified
//
#include <hip/hip_runtime.h>
#include <math.h>

typedef __attribute__((ext_vector_type(2))) float v2f;
typedef __attribute__((ext_vector_type(4))) float v4f;
typedef __attribute__((ext_vector_type(8))) float v8f;

static constexpr int Bn    = 4;
static constexpr int Sn    = 1024;
static constexpr int KNN   = 32;      // MAX_K
static constexpr int VOCAB = 32000;
static constexpr int HID   = 32;
static constexpr int RK    = 6;
static constexpr int IN    = 64;      // 2*MAX_K features
static constexpr int TOK   = Bn * Sn; // 4096
static constexpr int BLK_TOK = 64;    // tokens per block (2 waves x 2 M-tiles)
static constexpr int PAD   = IN + 2;  // LDS row pad: keeps 8B alignment, spreads banks

// fp32 WMMA: D(16x16,f32) = A(16x4,f32) x B(4x16,f32) + C
#define WMMA4F32(a, b, c) \
  __builtin_amdgcn_wmma_f32_16x16x4_f32(false, (a), false, (b), (short)0, (c), false, false)

// ---------------------------------------------------------------------------
// Kernel 0: stream-zero the 512MB output with non-temporal b128 stores.
// This is the roofline-dominant part of the whole problem (~44us @ 23.3TB/s).
// ---------------------------------------------------------------------------
__global__ void AC_zero_kernel(float* __restrict__ out, long long n) {
  long long n4 = n >> 2;
  long long i = (long long)blockIdx.x * blockDim.x + threadIdx.x;
  long long stride = (long long)gridDim.x * blockDim.x;
  v4f* __restrict__ o4 = (v4f*)out;
  v4f z = {};
  for (; i < n4; i += stride) __builtin_nontemporal_store(z, o4 + i);
  // scalar tail (out_size is divisible by 4 here, but stay generic)
  if (blockIdx.x == 0 && threadIdx.x == 0)
    for (long long t = n4 << 2; t < n; ++t) out[t] = 0.f;
}

// ---------------------------------------------------------------------------
// Kernel 1: per-token features -> WMMA MLPs -> blend -> scatter-add.
// Block = 64 threads (2 waves), 64 tokens. Wave w owns M-tiles 2w, 2w+1.
// ---------------------------------------------------------------------------
__global__ __launch_bounds__(BLK_TOK) void AC_combiner_kernel(
    const float* __restrict__ distances,
    const float* __restrict__ kW1, const float* __restrict__ kb1,
    const float* __restrict__ kW2, const float* __restrict__ kb2,
    const float* __restrict__ tW1, const float* __restrict__ tb1,
    const float* __restrict__ tW2, const float* __restrict__ tb2,
    const int*   __restrict__ vals,
    float*       __restrict__ out) {
  __shared__ int   s_vals[BLK_TOK][KNN];  // labels per token
  __shared__ float s_feat[BLK_TOK][PAD];  // features, later layer-1 hidden
  __shared__ float s_w1[64][PAD];         // [kW1 ; tW1] fused, row = output unit
  __shared__ float s_w2[16][PAD];         // [kW2|0 ; 0|tW2 ; zero pad] fused
  __shared__ float s_b1[64];
  __shared__ float s_b2[8];
  __shared__ float s_log[BLK_TOK][8];     // 6 k-logits + t-logit per token

  const int tid  = threadIdx.x;
  const int tok  = blockIdx.x * BLK_TOK + tid;  // one token per thread (phases A,C)

  // ---- stage fused weights / biases ----
  for (int i = tid; i < 64 * IN; i += BLK_TOK) {
    int o = i / IN, f = i % IN;
    s_w1[o][f] = (o < HID) ? kW1[o * IN + f] : tW1[(o - HID) * IN + f];
  }
  for (int i = tid; i < 16 * IN; i += BLK_TOK) {
    int r = i / IN, h = i % IN;
    float v = 0.f;
    if (r < RK)       { if (h < HID)  v = kW2[r * HID + h]; }
    else if (r == RK) { if (h >= HID) v = tW2[h - HID]; }
    s_w2[r][h] = v;
  }
  s_b1[tid] = (tid < HID) ? kb1[tid] : tb1[tid - HID];
  if (tid < 8) s_b2[tid] = (tid < RK) ? kb2[tid] : ((tid == RK) ? tb2[0] : 0.f);

  // ---- phase A: features (distances + distinct-nonzero-label prefix counts) ----
  for (int k = 0; k < KNN; ++k) {
    s_feat[tid][k]  = distances[(long long)tok * KNN + k];
    s_vals[tid][k]  = vals[(long long)tok * KNN + k];
  }
  {
    int cnt = 0;
    for (int j = 0; j < KNN; ++j) {
      int vj = s_vals[tid][j];
      int isnew = (vj != 0) ? 1 : 0;
      for (int m = 0; m < j; ++m) isnew &= (s_vals[tid][m] != vj) ? 1 : 0;
      cnt += isnew;
      s_feat[tid][KNN + j] = (float)cnt;
    }
  }
  __syncthreads();

  // ---- phase B: fused MLPs via fp32 WMMA ----
  const int wv   = tid >> 5;
  const int lane = tid & 31;
  const int lo   = lane & 15;   // M (A/D) or N (B/D) within tile
  const int hi   = lane >> 4;   // K-half selector / +8 row offset in D

  for (int t = 0; t < 2; ++t) {
    const int trow = (wv * 2 + t) * 16;  // first token row of this M-tile

    // layer 1: [16 x 64] @ [64 x 64]^T, 4 N-tiles x 16 K-steps
    v8f acc0 = {}, acc1 = {}, acc2 = {}, acc3 = {};
    for (int step = 0; step < 16; ++step) {
      const int f0 = step * 4 + hi * 2;  // A lane holds K = f0, f0+1
      v2f a;
      a.x = s_feat[trow + lo][f0];
      a.y = s_feat[trow + lo][f0 + 1];
      v2f b0, b1v, b2v, b3;
      b0.x  = s_w1[ 0 + lo][f0]; b0.y  = s_w1[ 0 + lo][f0 + 1];
      b1v.x = s_w1[16 + lo][f0]; b1v.y = s_w1[16 + lo][f0 + 1];
      b2v.x = s_w1[32 + lo][f0]; b2v.y = s_w1[32 + lo][f0 + 1];
      b3.x  = s_w1[48 + lo][f0]; b3.y  = s_w1[48 + lo][f0 + 1];
      acc0 = WMMA4F32(a, b0,  acc0);
      acc1 = WMMA4F32(a, b1v, acc1);
      acc2 = WMMA4F32(a, b2v, acc2);
      acc3 = WMMA4F32(a, b3,  acc3);
    }
    // bias + tanh; overwrite this tile's feature rows with hidden (wave-private)
    #pragma unroll
    for (int g = 0; g < 8; ++g) {
      const int row = trow + g + 8 * hi;
      s_feat[row][ 0 + lo] = tanhf(acc0[g] + s_b1[ 0 + lo]);
      s_feat[row][16 + lo] = tanhf(acc1[g] + s_b1[16 + lo]);
      s_feat[row][32 + lo] = tanhf(acc2[g] + s_b1[32 + lo]);
      s_feat[row][48 + lo] = tanhf(acc3[g] + s_b1[48 + lo]);
    }

    // layer 2: [16 x 64] @ [16 x 64]^T, one N-tile (cols 0..5 = k, col 6 = t)
    v8f accL2 = {};
    for (int step = 0; step < 16; ++step) {
      const int f0 = step * 4 + hi * 2;
      v2f a;
      a.x = s_feat[trow + lo][f0];
      a.y = s_feat[trow + lo][f0 + 1];
      v2f b;
      b.x = s_w2[lo][f0];
      b.y = s_w2[lo][f0 + 1];
      accL2 = WMMA4F32(a, b, accL2);
    }
    if (lo < 8) {
      #pragma unroll
      for (int g = 0; g < 8; ++g)
        s_log[trow + g + 8 * hi][lo] = accL2[g] + s_b2[lo];
    }
  }
  __syncthreads();

  // ---- phase C: per-token softmax / temperature / blend / scatter ----
  float kl[RK];
  float mx = -3.4e38f;
  #pragma unroll
  for (int r = 0; r < RK; ++r) { kl[r] = s_log[tid][r]; mx = fmaxf(mx, kl[r]); }
  float se = 0.f;
  #pragma unroll
  for (int r = 0; r < RK; ++r) { kl[r] = __expf(kl[r] - mx); se += kl[r]; }
  const float kinv = 1.f / se;
  const float temp = 1.f / (1.f + __expf(-s_log[tid][RK]));
  const float invT = 1.f / temp;

  float d[KNN], w[KNN], s[KNN];
  #pragma unroll
  for (int k = 0; k < KNN; ++k) {
    d[k] = distances[(long long)tok * KNN + k];
    w[k] = 0.f;
  }

  #pragma unroll
  for (int r = 0; r < RK; ++r) {
    const int lim = 1 << r;  // mask row 2^r - 1: factor 1 for k < lim, 1000 beyond
    float m2 = -3.4e38f;
    #pragma unroll
    for (int k = 0; k < KNN; ++k) {
      float sc = -d[k] * invT * ((k < lim) ? 1.f : 1000.f);
      s[k] = sc;
      m2 = fmaxf(m2, sc);
    }
    float ss = 0.f;
    #pragma unroll
    for (int k = 0; k < KNN; ++k) { float e = __expf(s[k] - m2); s[k] = e; ss += e; }
    const float p = kl[r] * kinv / ss;
    #pragma unroll
    for (int k = 0; k < KNN; ++k) w[k] += s[k] * p;
  }

  float* __restrict__ row = out + (long long)tok * VOCAB;
  #pragma unroll
  for (int k = 0; k < KNN; ++k)
    atomicAdd(row + s_vals[tid][k], w[k]);  // duplicates within a token accumulate
}

// ---------------------------------------------------------------------------
extern "C" void kernel_launch(void* const* d_in, const int* in_sizes, int n_in,
                              void* d_out, int out_size, void* d_ws, size_t ws_size,
                              hipStream_t stream) {
  (void)in_sizes; (void)n_in; (void)d_ws; (void)ws_size;
  const float* distances = (const float*)d_in[0];
  const float* kW1 = (const float*)d_in[1];
  const float* kb1 = (const float*)d_in[2];
  const float* kW2 = (const float*)d_in[3];
  const float* kb2 = (const float*)d_in[4];
  // d_in[5..8] = l_W1,l_b1,l_W2,l_b2 : lambda_ is unused by the reference output
  const float* tW1 = (const float*)d_in[9];
  const float* tb1 = (const float*)d_in[10];
  const float* tW2 = (const float*)d_in[11];
  const float* tb2 = (const float*)d_in[12];
  const int*   vals = (const int*)d_in[13];
  float* out = (float*)d_out;

  // 1) zero-fill the 512MB output (memory-bound dominant step)
  AC_zero_kernel<<<2048, 256, 0, stream>>>(out, (long long)out_size);

  // 2) fused feature/MLP/blend/scatter
  AC_combiner_kernel<<<TOK / BLK_TOK, BLK_TOK, 0, stream>>>(
      distances, kW1, kb1, kW2, kb2, tW1, tb1, tW2, tb2, vals, out);
}